// AttentionChannelPooling_58652073394237
// MI455X (gfx1250) — compile-verified
//
#include <hip/hip_runtime.h>
#include <math.h>

#define C_CH  512
#define B_N   8
#define MPAD  16      // WMMA M padding for batch
#define HW    16384
#define HDIM  1024
#define S_N   3
#define K_SEL 256
#define NBINS 2048

typedef __attribute__((ext_vector_type(2))) float v2f;
typedef __attribute__((ext_vector_type(8))) float v8f;

// order-preserving float -> uint key and inverse
__device__ __forceinline__ unsigned f2key(float f) {
    unsigned u = __float_as_uint(f);
    return (u & 0x80000000u) ? ~u : (u | 0x80000000u);
}
__device__ __forceinline__ float key2f(unsigned k) {
    unsigned u = (k & 0x80000000u) ? (k & 0x7fffffffu) : ~k;
    return __uint_as_float(u);
}

// ---------------------------------------------------------------------------
// Kernel 1: per-(b,c) channel stats: std, exact median, max.
// One 256-thread workgroup per channel. Uses 64KB LDS key buffer (CDNA5 WGP
// LDS is 320KB) + 2048-bin histogram for a 2-pass exact median selection.
// Output layout padded to [S, 16, C]; pad rows are never written; fc1 reads them as garbage.
// ---------------------------------------------------------------------------
__global__ __launch_bounds__(256) void stats_kernel(const float* __restrict__ x,
                                                    float* __restrict__ stats) {
    __shared__ unsigned hist[NBINS];     // 8 KB
    __shared__ unsigned keybuf[HW];      // 64 KB candidate keys
    __shared__ float s_sum, s_sumsq;
    __shared__ unsigned s_maxkey;
    __shared__ int s_cnt;                // candidate count
    __shared__ int s_zc;                 // per-bit zero count
    __shared__ int s_b0, s_b1, s_cumB0;

    const int bc  = blockIdx.x;          // b*512 + c
    const int tid = threadIdx.x;
    const float4* xp = (const float4*)(x + (size_t)bc * HW);

    for (int i = tid; i < NBINS; i += 256) hist[i] = 0u;
    if (tid == 0) { s_sum = 0.f; s_sumsq = 0.f; s_maxkey = 0u; s_cnt = 0; }
    __syncthreads();

    // ---- pass 1: sum / sumsq / max + top-11-bit histogram ----
    float lsum = 0.f, lsq = 0.f; unsigned lmax = 0u;
    for (int i = tid; i < HW / 4; i += 256) {
        float4 v = xp[i];
        float a[4] = {v.x, v.y, v.z, v.w};
#pragma unroll
        for (int j = 0; j < 4; ++j) {
            float f = a[j];
            lsum += f; lsq += f * f;
            unsigned k = f2key(f);
            if (k > lmax) lmax = k;
            atomicAdd(&hist[k >> 21], 1u);
        }
    }
    atomicAdd(&s_sum, lsum);
    atomicAdd(&s_sumsq, lsq);
    atomicMax(&s_maxkey, lmax);
    __syncthreads();

    // ---- locate bins containing order stats 8191 and 8192 ----
    if (tid == 0) {
        int cum = 0, b0 = -1, b1 = -1, cumB0 = 0;
        for (int i = 0; i < NBINS; ++i) {
            int c = (int)hist[i];
            if (b0 < 0 && cum + c > HW / 2 - 1) { b0 = i; cumB0 = cum; }
            if (b1 < 0 && cum + c > HW / 2)     { b1 = i; }
            cum += c;
        }
        s_b0 = b0; s_b1 = b1; s_cumB0 = cumB0;
    }
    __syncthreads();
    const int b0 = s_b0, b1 = s_b1, cumB0 = s_cumB0;

    // ---- pass 2: gather keys landing in the two candidate bins ----
    for (int i = tid; i < HW / 4; i += 256) {
        float4 v = xp[i];
        float a[4] = {v.x, v.y, v.z, v.w};
#pragma unroll
        for (int j = 0; j < 4; ++j) {
            unsigned k = f2key(a[j]);
            int bin = (int)(k >> 21);
            if (bin == b0 || bin == b1) {
                int p = atomicAdd(&s_cnt, 1);
                keybuf[p] = k;
            }
        }
    }
    __syncthreads();
    const int m = s_cnt;

    // ---- in-LDS bitwise radix select for the two middle ranks ----
    unsigned sel[2];
    const int targets[2] = {HW / 2 - 1 - cumB0, HW / 2 - cumB0};
#pragma unroll
    for (int r = 0; r < 2; ++r) {
        unsigned pref = 0u;
        int k = targets[r];
        for (int bit = 31; bit >= 0; --bit) {
            __syncthreads();
            if (tid == 0) s_zc = 0;
            __syncthreads();
            const int shift = bit + 1;
            int lc = 0;
            for (int i = tid; i < m; i += 256) {
                unsigned kk = keybuf[i];
                bool match = (shift >= 32) ? true : ((kk >> shift) == (pref >> shift));
                if (match && (((kk >> bit) & 1u) == 0u)) ++lc;
            }
            atomicAdd(&s_zc, lc);
            __syncthreads();
            int zc = s_zc;                 // uniform
            if (k >= zc) { k -= zc; pref |= (1u << bit); }
        }
        sel[r] = pref;
    }
    __syncthreads();

    if (tid == 0) {
        float mean = s_sum * (1.0f / HW);
        float var  = s_sumsq * (1.0f / HW) - mean * mean;
        float sd   = sqrtf(var > 0.f ? var : 0.f);
        float med  = 0.5f * (key2f(sel[0]) + key2f(sel[1]));
        float mx   = key2f(s_maxkey);
        const int b = bc / C_CH, c = bc % C_CH;          // [S,16,C] padded
        stats[(0 * MPAD + b) * C_CH + c] = sd;
        stats[(1 * MPAD + b) * C_CH + c] = med;
        stats[(2 * MPAD + b) * C_CH + c] = mx;
    }
}

// ---------------------------------------------------------------------------
// Kernel 2: h = relu(stats @ W1 + b1) via V_WMMA_F32_16X16X4_F32.
// One wave per (s, 16-wide N tile). M padded to 16: pad rows carry garbage
// but only influence pad rows of D, which are never read -> no predication.
// ---------------------------------------------------------------------------
__global__ __launch_bounds__(32) void fc1_kernel(const float* __restrict__ stats,
                                                 const float* __restrict__ W1,
                                                 const float* __restrict__ b1,
                                                 float* __restrict__ h) {
    const int s    = blockIdx.y;
    const int nt   = blockIdx.x;
    const int lane = threadIdx.x;
    const int m    = lane & 15;        // A row (both lane halves cover M=0..15)
    const int kg   = lane >> 4;        // K group: 0 -> K{0,1}, 1 -> K{2,3}
    const int n    = nt * 16 + (lane & 15);
    const float* A  = stats + (size_t)s * MPAD * C_CH;   // [16,512] padded
    const float* Bm = W1 + (size_t)s * C_CH * HDIM;      // [512,1024]

    v8f acc = {};
    for (int k0 = 0; k0 < C_CH; k0 += 4) {
        const int ka = k0 + 2 * kg;
        v2f a = *(const v2f*)(A + m * C_CH + ka);        // b64 load
        v2f b;
        b.x = Bm[(size_t)ka * HDIM + n];
        b.y = Bm[(size_t)(ka + 1) * HDIM + n];
        acc = __builtin_amdgcn_wmma_f32_16x16x4_f32(false, a, false, b,
                                                    (short)0, acc, false, false);
    }
    const float bias = b1[s * HDIM + n];
#pragma unroll
    for (int j = 0; j < 8; ++j) {
        int row = j + 8 * kg;          // C/D: VGPR j holds M=j (lanes 0-15), M=j+8 (16-31)
        float v = acc[j] + bias;
        h[((size_t)s * MPAD + row) * HDIM + n] = v > 0.f ? v : 0.f;
    }
}

// ---------------------------------------------------------------------------
// Kernel 3: logits = h @ W2 + b2 via V_WMMA_F32_16X16X4_F32 (padded M=16).
// ---------------------------------------------------------------------------
__global__ __launch_bounds__(32) void fc2_kernel(const float* __restrict__ h,
                                                 const float* __restrict__ W2,
                                                 const float* __restrict__ b2,
                                                 float* __restrict__ logits) {
    const int s    = blockIdx.y;
    const int nt   = blockIdx.x;
    const int lane = threadIdx.x;
    const int m    = lane & 15;
    const int kg   = lane >> 4;
    const int n    = nt * 16 + (lane & 15);
    const float* A  = h  + (size_t)s * MPAD * HDIM;      // [16,1024] padded
    const float* Bm = W2 + (size_t)s * HDIM * C_CH;      // [1024,512]

    v8f acc = {};
    for (int k0 = 0; k0 < HDIM; k0 += 4) {
        const int ka = k0 + 2 * kg;
        v2f a = *(const v2f*)(A + m * HDIM + ka);        // b64 load
        v2f b;
        b.x = Bm[(size_t)ka * C_CH + n];
        b.y = Bm[(size_t)(ka + 1) * C_CH + n];
        acc = __builtin_amdgcn_wmma_f32_16x16x4_f32(false, a, false, b,
                                                    (short)0, acc, false, false);
    }
    const float bias = b2[s * C_CH + n];
#pragma unroll
    for (int j = 0; j < 8; ++j) {
        int row = j + 8 * kg;
        logits[((size_t)s * MPAD + row) * C_CH + n] = acc[j] + bias;
    }
}

// ---------------------------------------------------------------------------
// Kernel 4: mean logits over S, rank-based top-256 per batch row.
// Softmax is monotonic -> ranking the mean logits gives identical ordering.
// Tie-break: equal values ordered by ascending index (lax.top_k semantics).
// ---------------------------------------------------------------------------
__global__ __launch_bounds__(512) void topk_kernel(const float* __restrict__ logits,
                                                   int* __restrict__ selIdx) {
    __shared__ float vals[C_CH];
    const int b = blockIdx.x;
    const int i = threadIdx.x;
    float v = (logits[(0 * MPAD + b) * C_CH + i] +
               logits[(1 * MPAD + b) * C_CH + i] +
               logits[(2 * MPAD + b) * C_CH + i]) * (1.0f / 3.0f);
    vals[i] = v;
    __syncthreads();
    int rank = 0;
    for (int j = 0; j < C_CH; ++j) {
        float vj = vals[j];
        rank += (vj > v) || (vj == v && j < i);
    }
    if (rank < K_SEL) selIdx[b * K_SEL + rank] = i;
}

// ---------------------------------------------------------------------------
// Kernel 5: gather selected channel maps (coalesced float4 copy, 256 MB).
// ---------------------------------------------------------------------------
__global__ __launch_bounds__(256) void gather_kernel(const float* __restrict__ x,
                                                     const int* __restrict__ selIdx,
                                                     float* __restrict__ out) {
    const int b = blockIdx.z;
    const int j = blockIdx.y;
    const int c = selIdx[b * K_SEL + j];
    const float4* src = (const float4*)(x   + ((size_t)(b * C_CH  + c)) * HW);
    float4*       dst = (float4*)      (out + ((size_t)(b * K_SEL + j)) * HW);
    const int idx = blockIdx.x * 256 + threadIdx.x;   // gridDim.x = HW/4/256 = 16
    dst[idx] = src[idx];
}

extern "C" void kernel_launch(void* const* d_in, const int* in_sizes, int n_in,
                              void* d_out, int out_size, void* d_ws, size_t ws_size,
                              hipStream_t stream) {
    const float* x  = (const float*)d_in[0];
    const float* W1 = (const float*)d_in[1];
    const float* b1 = (const float*)d_in[2];
    const float* W2 = (const float*)d_in[3];
    const float* b2 = (const float*)d_in[4];
    float* out = (float*)d_out;

    // padded workspace layouts [S,16,*]
    char* ws = (char*)d_ws;
    float* stats  = (float*)(ws);                             // 3*16*512  = 96 KB
    float* h      = (float*)(ws + 98304);                     // 3*16*1024 = 192 KB
    float* logits = (float*)(ws + 98304 + 196608);            // 3*16*512  = 96 KB
    int*   selIdx = (int*)  (ws + 98304 + 196608 + 98304);    // 8*256 ints

    stats_kernel<<<B_N * C_CH, 256, 0, stream>>>(x, stats);
    fc1_kernel<<<dim3(HDIM / 16, S_N), 32, 0, stream>>>(stats, W1, b1, h);
    fc2_kernel<<<dim3(C_CH / 16, S_N), 32, 0, stream>>>(h, W2, b2, logits);
    topk_kernel<<<B_N, C_CH, 0, stream>>>(logits, selIdx);
    gather_kernel<<<dim3(HW / 4 / 256, K_SEL, B_N), 256, 0, stream>>>(x, selIdx, out);
}